// nUIV_NODE_8134668058829
// MI455X (gfx1250) — compile-verified
//
#include <hip/hip_runtime.h>
#include <math.h>

#define N_NODES   4096
#define NQ4       1024          // N/4 in float4 units
#define T_STEPS   64
#define NBLOCKS   128
#define NTHREADS  256
#define MSLICES   4             // 128 blocks = 32 n-tiles x 4 m-slices
#define NTILES    32

typedef float v2f __attribute__((ext_vector_type(2)));
typedef float v8f __attribute__((ext_vector_type(8)));

// workspace float-offsets (after 64-byte barrier header)
#define OFF_U     0
#define OFF_I     (1*N_NODES)
#define OFF_V     (2*N_NODES)
#define OFF_US    (3*N_NODES)
#define OFF_IS    (4*N_NODES)
#define OFF_VS    (5*N_NODES)
#define OFF_AU    (6*N_NODES)
#define OFF_AI    (7*N_NODES)
#define OFF_AV    (8*N_NODES)
#define OFF_PART  (9*N_NODES)            // MSLICES * N
#define OFF_DIAG  (13*N_NODES)
#define OFF_B2    (14*N_NODES)
#define OFF_D2    (15*N_NODES)
#define OFF_P2    (16*N_NODES)
#define OFF_C2    (17*N_NODES)
#define OFF_BSUM  (18*N_NODES)           // 16 blocks * 3
#define OFF_SIG   (18*N_NODES + 64)      // T*3 = 192

__device__ __forceinline__ float sigmoidf(float x) {
    return 1.0f / (1.0f + expf(-x));
}

// ---- CDNA5 async copy: global -> LDS (ASYNCcnt-tracked DMA, no VGPR round trip)
__device__ __forceinline__ void async_tile_load_b128(unsigned ldsoff, const void* gaddr) {
    asm volatile("global_load_async_to_lds_b128 %0, %1, off"
                 :: "v"(ldsoff), "v"(gaddr)
                 : "memory");
}

#if defined(__has_builtin)
#if __has_builtin(__builtin_amdgcn_s_wait_asynccnt)
#define WAIT_ASYNCCNT(n) __builtin_amdgcn_s_wait_asynccnt(n)
#endif
#endif
#ifndef WAIT_ASYNCCNT
#define WAIT_ASYNCCNT(n) asm volatile("s_wait_asynccnt %0" :: "n"(n) : "memory")
#endif

// low 32 bits of a flat shared-memory address are the LDS byte offset
__device__ __forceinline__ unsigned lds_off32(const void* p) {
    return (unsigned)(uintptr_t)p;
}

// Device-wide sense-reversal barrier (persistent-kernel grid sync).
__device__ __forceinline__ void gsync(unsigned* cnt, unsigned* gen) {
    __threadfence();            // make this thread's global writes device-visible
    __syncthreads();
    if (threadIdx.x == 0) {
        unsigned g = __hip_atomic_load(gen, __ATOMIC_RELAXED, __HIP_MEMORY_SCOPE_AGENT);
        unsigned prev = __hip_atomic_fetch_add(cnt, 1u, __ATOMIC_ACQ_REL, __HIP_MEMORY_SCOPE_AGENT);
        if (prev == (unsigned)(NBLOCKS - 1)) {
            __hip_atomic_store(cnt, 0u, __ATOMIC_RELAXED, __HIP_MEMORY_SCOPE_AGENT);
            __hip_atomic_fetch_add(gen, 1u, __ATOMIC_RELEASE, __HIP_MEMORY_SCOPE_AGENT);
        } else {
            while (__hip_atomic_load(gen, __ATOMIC_ACQUIRE, __HIP_MEMORY_SCOPE_AGENT) == g)
                __builtin_amdgcn_s_sleep(2);
        }
    }
    __syncthreads();
    __threadfence();            // acquire side: discard stale near caches
}

// Block-level 3-component sigmoid sum into bsum[blockIdx*3 + {0,1,2}]
__device__ __forceinline__ void sig_reduce_block(float u, float i, float v,
                                                 const float* __restrict__ slope,
                                                 const float* __restrict__ thr,
                                                 float* __restrict__ bsum,
                                                 float* redu, float* redi, float* redv) {
    redu[threadIdx.x] = sigmoidf(slope[0] * (u - thr[0]));
    redi[threadIdx.x] = sigmoidf(slope[1] * (i - thr[1]));
    redv[threadIdx.x] = sigmoidf(slope[2] * (v - thr[2]));
    __syncthreads();
    for (int off = NTHREADS / 2; off > 0; off >>= 1) {
        if (threadIdx.x < (unsigned)off) {
            redu[threadIdx.x] += redu[threadIdx.x + off];
            redi[threadIdx.x] += redi[threadIdx.x + off];
            redv[threadIdx.x] += redv[threadIdx.x + off];
        }
        __syncthreads();
    }
    if (threadIdx.x == 0) {
        bsum[blockIdx.x * 3 + 0] = redu[0];
        bsum[blockIdx.x * 3 + 1] = redi[0];
        bsum[blockIdx.x * 3 + 2] = redv[0];
    }
}

__global__ __launch_bounds__(NTHREADS, 1)
void nuiv_persistent(const float* __restrict__ times,
                     const float* __restrict__ x0,
                     const float* __restrict__ betas,
                     const float* __restrict__ deltas,
                     const float* __restrict__ cs,
                     const float* __restrict__ ps,
                     const float* __restrict__ ts,
                     const float* __restrict__ W,
                     const float* __restrict__ bvec,
                     const float* __restrict__ slope,
                     const float* __restrict__ thr,
                     float* __restrict__ out,
                     void* __restrict__ wsraw) {
    unsigned* cnt = (unsigned*)wsraw;
    unsigned* gen = cnt + 1;
    float* wsf = (float*)wsraw + 16;      // skip 64-byte header

    float* U    = wsf + OFF_U;
    float* I    = wsf + OFF_I;
    float* V    = wsf + OFF_V;
    float* Us   = wsf + OFF_US;
    float* Is   = wsf + OFF_IS;
    float* Vs   = wsf + OFF_VS;
    float* aU   = wsf + OFF_AU;
    float* aI   = wsf + OFF_AI;
    float* aV   = wsf + OFF_AV;
    float* part = wsf + OFF_PART;
    float* diag = wsf + OFF_DIAG;
    float* b2   = wsf + OFF_B2;
    float* d2   = wsf + OFF_D2;
    float* p2   = wsf + OFF_P2;
    float* c2   = wsf + OFF_C2;
    float* bsum = wsf + OFF_BSUM;
    float* sgsm = wsf + OFF_SIG;

    // LDS: double-buffered 32x(128+4 pad) f32 ts tile + 1024 f32 V slice + reduce scratch
    __shared__ float4 tile4s[2][32 * 33];
    __shared__ float  vlds[1024];
    __shared__ float  redu[NTHREADS], redi[NTHREADS], redv[NTHREADS];

    const int gtid   = blockIdx.x * NTHREADS + threadIdx.x;
    const int lane   = threadIdx.x & 31;
    const int wv     = threadIdx.x >> 5;          // wave id in block, 0..7
    const int half   = lane >> 4;                 // 0 = lanes 0-15, 1 = lanes 16-31
    const int jloc   = wv * 16 + (lane & 15);     // 0..127: n offset inside tile
    const int ntile  = blockIdx.x & (NTILES - 1);
    const int mslice = blockIdx.x >> 5;           // 0..3
    const int n0     = ntile * 128;
    const int mbase  = mslice * 1024;
    const float4* ts4 = (const float4*)ts;

    // per-thread staging geometry: rows wv + 8*k2, float4-column = lane
    const unsigned ldsbase0 = lds_off32(&tile4s[0][0]);
    const unsigned ldsbase1 = lds_off32(&tile4s[1][0]);

    // issue the 4 async B128 copies this thread owns for one 32x128 chunk
    auto issue_chunk = [&](int chunk, int buf) {
        const int mrow0 = mbase + chunk * 32;
        const unsigned lbase = buf ? ldsbase1 : ldsbase0;
        #pragma unroll
        for (int k2 = 0; k2 < 4; ++k2) {
            int mm = wv + 8 * k2;
            unsigned loff = lbase + (unsigned)(mm * 33 + lane) * 16u;
            const float4* g = ts4 + (((size_t)(mrow0 + mm)) << 10) + (n0 >> 2) + lane;
            async_tile_load_b128(loff, (const void*)g);
        }
    };

    // ---------------- init: y0 = x0^2, params, diag, sigmoid(t=0) ----------------
    {
        float u = 0.f, iy = 0.f, v = 0.f;
        if (gtid < N_NODES) {
            int n = gtid;
            u  = x0[3 * n + 0]; u  *= u;
            iy = x0[3 * n + 1]; iy *= iy;
            v  = x0[3 * n + 2]; v  *= v;
            U[n] = u;  I[n] = iy;  V[n] = v;
            Us[n] = u; Is[n] = iy; Vs[n] = v;
            diag[n] = ts[(size_t)n * N_NODES + n];
            float t;
            t = betas[n];  b2[n] = t * t;
            t = deltas[n]; d2[n] = t * t;
            t = ps[n];     p2[n] = t * t;
            t = cs[n];     c2[n] = t * t;
        }
        if (blockIdx.x < 16)
            sig_reduce_block(u, iy, v, slope, thr, bsum, redu, redi, redv);
    }
    gsync(cnt, gen);

    // ---------------- time loop: 63 RK4 steps ----------------
    for (int step = 0; step < T_STEPS - 1; ++step) {
        // fold previous bsum (state at time index `step`) into sigsum
        if (blockIdx.x == 0 && threadIdx.x < 3) {
            float s = 0.f;
            for (int bb = 0; bb < 16; ++bb) s += bsum[bb * 3 + threadIdx.x];
            sgsm[step * 3 + threadIdx.x] = s;
        }
        const float dtv = times[step + 1] - times[step];

        for (int s = 0; s < 4; ++s) {
            // ======== mat-vec phase: part[mslice][n] = sum_m ts[m][n] * Vs[m] ========
            {
                // stage this block's V slice (1024 values) into LDS
                for (int k2 = 0; k2 < 4; ++k2)
                    vlds[k2 * 256 + threadIdx.x] = Vs[mbase + k2 * 256 + threadIdx.x];

                v8f c = {};   // 16x16 f32 accumulator (rows replicated by construction)
                issue_chunk(0, 0);              // prologue: DMA chunk 0 into buf 0
                for (int chunk = 0; chunk < 32; ++chunk) {
                    const int cur = chunk & 1;
                    if (chunk + 1 < 32) {
                        issue_chunk(chunk + 1, cur ^ 1);   // overlap next chunk's DMA
                        WAIT_ASYNCCNT(4);                  // drain current chunk's 4 copies
                    } else {
                        WAIT_ASYNCCNT(0);
                    }
                    __syncthreads();            // whole tile resident for all waves
                    const float* tilef = (const float*)tile4s[cur];
                    const int mc = chunk * 32;
                    #pragma unroll
                    for (int mm = 0; mm < 32; mm += 4) {
                        // A (16x4, M-replicated broadcast of V): lanes<16 K={0,1}, lanes>=16 K={2,3}
                        v2f a;
                        int mr = mc + mm + 2 * half;
                        a[0] = vlds[mr];
                        a[1] = vlds[mr + 1];
                        // B (4x16 ts tile): VGPR0 rows {0,2}, VGPR1 rows {1,3}
                        v2f b;
                        int r0 = mm + 2 * half;
                        b[0] = tilef[r0 * 132 + jloc];
                        b[1] = tilef[(r0 + 1) * 132 + jloc];
                        c = __builtin_amdgcn_wmma_f32_16x16x4_f32(
                                false, a, false, b, (short)0, c, false, false);
                    }
                    __syncthreads();            // all waves done reading buf[cur]
                }
                // every row of C is identical; row 0 cols live in c[0] lanes 0..15
                if (lane < 16)
                    part[mslice * N_NODES + n0 + wv * 16 + lane] = c[0];
            }
            gsync(cnt, gen);

            // ======== combine phase: RK4 stage update (blocks 0..15 active) ========
            {
                float nu = 0.f, ni = 0.f, nv = 0.f;
                if (gtid < N_NODES) {
                    int n = gtid;
                    float vold = Vs[n];
                    float cp = part[n] + part[N_NODES + n] + part[2 * N_NODES + n]
                             + part[3 * N_NODES + n] - diag[n] * vold;
                    float uold = Us[n], iold = Is[n];
                    float infect = b2[n] * uold * vold;
                    float kU = -infect;
                    float kI = infect - d2[n] * iold;
                    float kV = p2[n] * iold - c2[n] * vold + cp;
                    if (s == 0)      { aU[n] = kU; aI[n] = kI; aV[n] = kV; }
                    else if (s < 3)  { aU[n] += 2.f * kU; aI[n] += 2.f * kI; aV[n] += 2.f * kV; }
                    if (s < 3) {
                        float f = (s == 2) ? dtv : 0.5f * dtv;
                        Us[n] = U[n] + f * kU;
                        Is[n] = I[n] + f * kI;
                        Vs[n] = V[n] + f * kV;
                    } else {
                        float w6 = dtv * (1.0f / 6.0f);
                        nu = U[n] + w6 * (aU[n] + kU);
                        ni = I[n] + w6 * (aI[n] + kI);
                        nv = V[n] + w6 * (aV[n] + kV);
                        U[n] = nu;  I[n] = ni;  V[n] = nv;
                        Us[n] = nu; Is[n] = ni; Vs[n] = nv;
                    }
                }
                if (s == 3 && blockIdx.x < 16)
                    sig_reduce_block(nu, ni, nv, slope, thr, bsum, redu, redi, redv);
            }
            gsync(cnt, gen);
        }
    }

    // fold final bsum (t = T-1)
    if (blockIdx.x == 0 && threadIdx.x < 3) {
        float s = 0.f;
        for (int bb = 0; bb < 16; ++bb) s += bsum[bb * 3 + threadIdx.x];
        sgsm[(T_STEPS - 1) * 3 + threadIdx.x] = s;
    }
    gsync(cnt, gen);

    // ---------------- output: SIR[i][t] = b[i] + sum_j W[i,j] * S_j[t] / N ----------------
    if (blockIdx.x == 0 && threadIdx.x < 3 * T_STEPS) {
        int idx = threadIdx.x;
        int i = idx / T_STEPS;
        int t = idx % T_STEPS;
        float val = W[i * 3 + 0] * sgsm[t * 3 + 0]
                  + W[i * 3 + 1] * sgsm[t * 3 + 1]
                  + W[i * 3 + 2] * sgsm[t * 3 + 2];
        out[idx] = bvec[i] + val * (1.0f / (float)N_NODES);
    }
}

extern "C" void kernel_launch(void* const* d_in, const int* in_sizes, int n_in,
                              void* d_out, int out_size, void* d_ws, size_t ws_size,
                              hipStream_t stream) {
    (void)in_sizes; (void)n_in; (void)out_size; (void)ws_size;
    // zero the grid-barrier header (d_ws is poisoned by the harness)
    hipMemsetAsync(d_ws, 0, 64, stream);
    nuiv_persistent<<<NBLOCKS, NTHREADS, 0, stream>>>(
        (const float*)d_in[0],   // times
        (const float*)d_in[1],   // x0
        (const float*)d_in[2],   // betas
        (const float*)d_in[3],   // deltas
        (const float*)d_in[4],   // cs
        (const float*)d_in[5],   // ps
        (const float*)d_in[6],   // ts
        (const float*)d_in[7],   // W
        (const float*)d_in[8],   // b
        (const float*)d_in[9],   // slope
        (const float*)d_in[10],  // threshold
        (float*)d_out,
        d_ws);
}